// TPAttention_35579509080079
// MI455X (gfx1250) — compile-verified
//
#include <hip/hip_runtime.h>
#include <hip/hip_bf16.h>

typedef __attribute__((ext_vector_type(2))) float v2f;
typedef __attribute__((ext_vector_type(8))) float v8f;

#define BQ 8
#define SQ 4
#define PP 4096
#define HID 4096
#define HQ 32
#define HKV 8
#define DD 128
#define TT 4100          // P + S
#define NSPLIT 17        // ceil(4100 / 256)
#define SPLIT_KEYS 256
#define GEMM_KC 32       // K-chunk staged in LDS

// ---------------------------------------------------------------------------
// Skinny GEMM: C(32xN) = A(32xK) @ B(KxN), fp32, V_WMMA_F32_16X16X4_F32.
// Block = 128 threads = 4 waves = 64 output columns.
// B chunk (GEMM_KC x 64) staged in LDS via coalesced float4 loads; the WMMA
// k-loop then reads operands from LDS at immediate offsets.
// ---------------------------------------------------------------------------
__global__ __launch_bounds__(128) void gemm32_wmma(
    const float* __restrict__ A, const float* __restrict__ Bw,
    float* __restrict__ C, int K, int N) {
  __shared__ float Bs[GEMM_KC * 64];
  const int tid  = threadIdx.x;
  const int wave = tid >> 5;
  const int lane = tid & 31;
  const int nb   = blockIdx.x * 64;
  const int ml   = lane & 15;
  const int kboff = (lane >> 4) * 2;   // K-half per lane-half
  const int wn   = wave * 16;

  const float* ap0 = A + (size_t)ml * K + kboff;          // rows 0..15
  const float* ap1 = A + (size_t)(ml + 16) * K + kboff;   // rows 16..31

  v8f c0 = {}; v8f c1 = {};
  for (int k0 = 0; k0 < K; k0 += GEMM_KC) {
    // stage B chunk to LDS: 32 rows x 64 cols = 512 float4, 4 per thread
#pragma unroll
    for (int it = 0; it < 4; ++it) {
      int i  = tid + it * 128;         // 0..511
      int r  = i >> 4;                 // 0..31
      int c4 = (i & 15) * 4;
      *(float4*)(Bs + r * 64 + c4) =
          *(const float4*)(Bw + (size_t)(k0 + r) * N + nb + c4);
    }
    __syncthreads();
#pragma unroll
    for (int k = 0; k < GEMM_KC; k += 4) {
      const int kk = k + kboff;
      v2f a0 = *(const v2f*)(ap0 + k0 + k);
      v2f a1 = *(const v2f*)(ap1 + k0 + k);
      v2f bb;
      bb.x = Bs[kk * 64 + wn + ml];
      bb.y = Bs[(kk + 1) * 64 + wn + ml];
      c0 = __builtin_amdgcn_wmma_f32_16x16x4_f32(false, a0, false, bb, (short)0, c0, false, false);
      c1 = __builtin_amdgcn_wmma_f32_16x16x4_f32(false, a1, false, bb, (short)0, c1, false, false);
    }
    __syncthreads();
  }
  const int rh = (lane >> 4) * 8;
  const int n  = nb + wn + ml;
#pragma unroll
  for (int v = 0; v < 8; ++v) {
    C[(size_t)(v + rh) * N + n]      = c0[v];
    C[(size_t)(v + rh + 16) * N + n] = c1[v];
  }
}

// ---------------------------------------------------------------------------
// RoPE on q (32x32x128) and k (32x8x128); v passes through untouched.
// ---------------------------------------------------------------------------
__global__ void rope_kernel(const float* __restrict__ qraw,
                            const float* __restrict__ kraw,
                            const float* __restrict__ cosb,
                            const float* __restrict__ sinb,
                            const int*   __restrict__ pastp,
                            float* __restrict__ qr, float* __restrict__ kr) {
  const int idx  = blockIdx.x * blockDim.x + threadIdx.x;
  const int past = pastp[0];
  const int NQ = BQ * SQ * HQ * DD;   // 131072
  const int NK = BQ * SQ * HKV * DD;  // 32768
  if (idx < NQ) {
    int d = idx & 127;
    int row = idx >> 12;              // /(HQ*DD)
    int s = row & 3;
    int pos = past + s;
    float c  = cosb[pos * DD + d];
    float sn = sinb[pos * DD + d];
    float x  = qraw[idx];
    float rot = (d < 64) ? -qraw[idx + 64] : qraw[idx - 64];
    qr[idx] = x * c + rot * sn;
  } else if (idx < NQ + NK) {
    int j = idx - NQ;
    int d = j & 127;
    int row = j >> 10;                // /(HKV*DD)
    int s = row & 3;
    int pos = past + s;
    float c  = cosb[pos * DD + d];
    float sn = sinb[pos * DD + d];
    float x  = kraw[j];
    float rot = (d < 64) ? -kraw[j + 64] : kraw[j - 64];
    kr[j] = x * c + rot * sn;
  }
}

// ---------------------------------------------------------------------------
// Flash-decoding attention: one 32-thread wave per (b, kv, split).
// 16 query rows (m = s*4 + g), split handles 256 keys in 16-key chunks.
// ---------------------------------------------------------------------------
__global__ __launch_bounds__(32) void attn_kernel(
    const float* __restrict__ qr, const float* __restrict__ kr,
    const float* __restrict__ vr,
    const float* __restrict__ kc, const float* __restrict__ vc,
    float* __restrict__ pO, float* __restrict__ pM, float* __restrict__ pL) {
  __shared__ float Qs[16 * 128];
  __shared__ float Ks[16 * 128];
  __shared__ float Vs[16 * 128];
  __shared__ float Ps[16 * 16];

  const int lane  = threadIdx.x;
  const int split = blockIdx.x % NSPLIT;
  const int pair  = blockIdx.x / NSPLIT;   // 0..63
  const int kv = pair & 7;
  const int b  = pair >> 3;

  const float scale = 0.08838834764831845f;  // 1/sqrt(128)
  // stage Q (16x128) as float4, pre-scaled
  for (int i = lane; i < 512; i += 32) {
    int m = i >> 5, d4 = (i & 31) * 4;
    int s = m >> 2, g = m & 3;
    float4 qv = *(const float4*)(qr +
        ((size_t)((b * SQ + s) * HQ) + kv * 4 + g) * DD + d4);
    qv.x *= scale; qv.y *= scale; qv.z *= scale; qv.w *= scale;
    *(float4*)(Qs + m * 128 + d4) = qv;
  }

  v8f O[8];
  float mrow[8], lrow[8];
#pragma unroll
  for (int dt = 0; dt < 8; ++dt) O[dt] = (v8f){0,0,0,0,0,0,0,0};
#pragma unroll
  for (int v = 0; v < 8; ++v) { mrow[v] = -1e30f; lrow[v] = 0.f; }

  const int t_begin = split * SPLIT_KEYS;
  int t_end = t_begin + SPLIT_KEYS; if (t_end > TT) t_end = TT;
  const int ml    = lane & 15;
  const int half  = lane >> 4;
  const int kboff = half * 2;
  __syncthreads();

  for (int t0 = t_begin; t0 < t_end; t0 += 16) {
    // stage K,V chunk (16 rows x 128) into LDS, float4-coalesced
    for (int i = lane; i < 512; i += 32) {
      int r = i >> 5, d4 = (i & 31) * 4;
      int tk = t0 + r;
      float4 kvl{}; float4 vvl{};
      if (tk < PP) {
        size_t off = (((size_t)b * PP + tk) * HKV + kv) * DD + d4;
        kvl = *(const float4*)(kc + off);
        vvl = *(const float4*)(vc + off);
      } else if (tk < TT) {
        int s = tk - PP;
        size_t off = ((size_t)(b * SQ + s) * HKV + kv) * DD + d4;
        kvl = *(const float4*)(kr + off);
        vvl = *(const float4*)(vr + off);
      }
      *(float4*)(Ks + r * 128 + d4) = kvl;
      *(float4*)(Vs + r * 128 + d4) = vvl;
    }
    __syncthreads();

    // scores: Q(16x128) @ K^T(128x16)
    v8f sc = (v8f){0,0,0,0,0,0,0,0};
#pragma unroll
    for (int k = 0; k < 128; k += 4) {
      const int kk = k + kboff;
      v2f a  = *(const v2f*)(Qs + ml * 128 + kk);
      v2f bb = *(const v2f*)(Ks + ml * 128 + kk);
      sc = __builtin_amdgcn_wmma_f32_16x16x4_f32(false, a, false, bb, (short)0, sc, false, false);
    }
    // causal + bounds mask (t <= past + s covers t < 4100 too)
#pragma unroll
    for (int v = 0; v < 8; ++v) {
      int row = v + 8 * half;
      int sq  = row >> 2;
      int tk  = t0 + ml;
      if (tk > PP + sq) sc[v] = -1e30f;
    }
    // row max (butterfly within 16-lane half)
    float rmax[8];
#pragma unroll
    for (int v = 0; v < 8; ++v) rmax[v] = sc[v];
    for (int off = 1; off < 16; off <<= 1) {
#pragma unroll
      for (int v = 0; v < 8; ++v)
        rmax[v] = fmaxf(rmax[v], __shfl_xor(rmax[v], off, 32));
    }
    float mscale[8];
#pragma unroll
    for (int v = 0; v < 8; ++v) {
      float mnew = fmaxf(mrow[v], rmax[v]);
      mscale[v] = __expf(mrow[v] - mnew);
      mrow[v]   = mnew;
      sc[v]     = __expf(sc[v] - mnew);
    }
    // row sum
    float rsum[8];
#pragma unroll
    for (int v = 0; v < 8; ++v) rsum[v] = sc[v];
    for (int off = 1; off < 16; off <<= 1) {
#pragma unroll
      for (int v = 0; v < 8; ++v)
        rsum[v] += __shfl_xor(rsum[v], off, 32);
    }
#pragma unroll
    for (int v = 0; v < 8; ++v) lrow[v] = lrow[v] * mscale[v] + rsum[v];
    // rescale accumulators
#pragma unroll
    for (int dt = 0; dt < 8; ++dt)
#pragma unroll
      for (int v = 0; v < 8; ++v) O[dt][v] *= mscale[v];

    // P: C-layout -> A-layout via LDS
    __syncthreads();
#pragma unroll
    for (int v = 0; v < 8; ++v) Ps[(v + 8 * half) * 16 + ml] = sc[v];
    __syncthreads();

    // O += P(16x16) @ V(16x128)
#pragma unroll
    for (int k = 0; k < 16; k += 4) {
      const int kk = k + kboff;
      v2f a = *(const v2f*)(Ps + ml * 16 + kk);
#pragma unroll
      for (int dt = 0; dt < 8; ++dt) {
        v2f bb;
        bb.x = Vs[kk * 128 + dt * 16 + ml];
        bb.y = Vs[(kk + 1) * 128 + dt * 16 + ml];
        O[dt] = __builtin_amdgcn_wmma_f32_16x16x4_f32(false, a, false, bb, (short)0, O[dt], false, false);
      }
    }
    __syncthreads();
  }

  // write partials
  const size_t base = ((size_t)pair * NSPLIT + split) * 16;
#pragma unroll
  for (int dt = 0; dt < 8; ++dt)
#pragma unroll
    for (int v = 0; v < 8; ++v) {
      int row = v + 8 * half;
      pO[(base + row) * 128 + dt * 16 + ml] = O[dt][v];
    }
  if (ml == 0) {
#pragma unroll
    for (int v = 0; v < 8; ++v) {
      pM[base + v + 8 * half] = mrow[v];
      pL[base + v + 8 * half] = lrow[v];
    }
  }
}

// ---------------------------------------------------------------------------
// Log-sum-exp merge of the T-splits.
// ---------------------------------------------------------------------------
__global__ void reduce_kernel(const float* __restrict__ pO,
                              const float* __restrict__ pM,
                              const float* __restrict__ pL,
                              float* __restrict__ aout) {
  const int idx = blockIdx.x * blockDim.x + threadIdx.x;  // 64*16*128
  if (idx >= 64 * 16 * 128) return;
  const int d    = idx & 127;
  const int row  = (idx >> 7) & 15;
  const int pair = idx >> 11;
  const int kv = pair & 7, b = pair >> 3;
  const int sq = row >> 2, g = row & 3;

  float mstar = -1e30f;
  for (int p = 0; p < NSPLIT; ++p)
    mstar = fmaxf(mstar, pM[((size_t)pair * NSPLIT + p) * 16 + row]);
  float lsum = 0.f, acc = 0.f;
  for (int p = 0; p < NSPLIT; ++p) {
    size_t base = ((size_t)pair * NSPLIT + p) * 16 + row;
    float w = __expf(pM[base] - mstar);
    lsum += pL[base] * w;
    acc  += pO[base * 128 + d] * w;
  }
  aout[((size_t)((b * SQ + sq) * HQ) + kv * 4 + g) * DD + d] = acc / lsum;
}

// ---------------------------------------------------------------------------
// Launcher
// ---------------------------------------------------------------------------
extern "C" void kernel_launch(void* const* d_in, const int* in_sizes, int n_in,
                              void* d_out, int out_size, void* d_ws, size_t ws_size,
                              hipStream_t stream) {
  const float* x  = (const float*)d_in[0];
  const float* wq = (const float*)d_in[1];
  const float* wk = (const float*)d_in[2];
  const float* wv = (const float*)d_in[3];
  const float* wo = (const float*)d_in[4];
  const float* cosb = (const float*)d_in[5];
  const float* sinb = (const float*)d_in[6];
  const float* kc = (const float*)d_in[7];
  const float* vc = (const float*)d_in[8];
  const int*   pastp = (const int*)d_in[9];
  float* out = (float*)d_out;

  float* ws = (float*)d_ws;
  float* q_raw = ws;                     // 32*4096
  float* k_raw = q_raw + 131072;         // 32*1024
  float* v_raw = k_raw + 32768;          // 32*1024
  float* q_r   = v_raw + 32768;          // 32*4096
  float* k_r   = q_r + 131072;           // 32*1024
  float* pO    = k_r + 32768;            // 64*17*16*128
  float* pM    = pO + 64 * NSPLIT * 16 * 128;
  float* pL    = pM + 64 * NSPLIT * 16;
  float* aout  = pL + 64 * NSPLIT * 16;  // 32*4096

  // QKV projections
  gemm32_wmma<<<64, 128, 0, stream>>>(x, wq, q_raw, HID, HQ * DD);
  gemm32_wmma<<<16, 128, 0, stream>>>(x, wk, k_raw, HID, HKV * DD);
  gemm32_wmma<<<16, 128, 0, stream>>>(x, wv, v_raw, HID, HKV * DD);
  // RoPE
  rope_kernel<<<(131072 + 32768 + 255) / 256, 256, 0, stream>>>(
      q_raw, k_raw, cosb, sinb, pastp, q_r, k_r);
  // attention (flash-decoding over 17 T-splits)
  attn_kernel<<<64 * NSPLIT, 32, 0, stream>>>(q_r, k_r, v_raw, kc, vc, pO, pM, pL);
  reduce_kernel<<<(64 * 16 * 128 + 255) / 256, 256, 0, stream>>>(pO, pM, pL, aout);
  // output projection
  gemm32_wmma<<<64, 128, 0, stream>>>(aout, wo, out, HID, HID);
}